// MANEAttention_9938554323613
// MI455X (gfx1250) — compile-verified
//
#include <hip/hip_runtime.h>
#include <hip/hip_bf16.h>

typedef __attribute__((ext_vector_type(2))) float v2f;
typedef __attribute__((ext_vector_type(8))) float v8f;
typedef __attribute__((ext_vector_type(4))) unsigned int v4u;
typedef __attribute__((ext_vector_type(8))) int v8i;
typedef __attribute__((ext_vector_type(4))) int v4i;

#define VV     3
#define NN     200000
#define DD     128
#define BB     32768
#define KK     10
#define TOT    65536
#define TILES  (BB / 16)     /* 2048 tiles of 16 rows */
#define TERMS  15

__device__ __forceinline__ float log_sigmoid(float x) {
    // jax.nn.log_sigmoid(x) = min(x,0) - log1p(exp(-|x|))
    return fminf(x, 0.0f) - log1pf(__expf(-fabsf(x)));
}

__device__ __forceinline__ unsigned int lds_byte_off(void* p) {
    // generic -> LDS(AS3) addrspacecast, then ptrtoint: offset in wave's LDS
    return (unsigned int)(uintptr_t)(__attribute__((address_space(3))) void*)p;
}

// One TDM gather-mode op: pull 8 rows of 128 f32 (4 KB) from `tab` into LDS.
// D# per CDNA5 ISA:
//   g0: count=1 | gather_index_size(32b)<<30 | gather_mode<<31 ; lds_addr ;
//       global_addr[56:0] ; type=2
//   g1: data_size=4B ; tensor_dim0=128 ; tensor_dim1=N ; tile_dim0=128 ;
//       tile_dim1=8 valid indices ; tensor_dim0_stride=128
//   g2/g3: row_index_0..3 / 4..7 (32-bit); 5th operand carries all 8 indices.
__device__ __forceinline__ void tdm_gather8(unsigned int lds_addr,
                                            const float* tab,
                                            v4i idx_lo, v4i idx_hi) {
    unsigned long long ga = (unsigned long long)(uintptr_t)tab;
    v4u g0;
    g0.x = 0xC0000001u;                 // count=1, gather_mode, 32-bit indices
    g0.y = lds_addr;
    g0.z = (unsigned int)ga;
    g0.w = ((unsigned int)(ga >> 32) & 0x01FFFFFFu) | 0x80000000u; // type=2
    v8i g1;
    g1[0] = 0x20000;                              // data_size = 2 (4 bytes)
    g1[1] = (int)(128u << 16);                    // tensor_dim0[15:0] = 128
    g1[2] = (int)(((unsigned)NN & 0xFFFFu) << 16);// tensor_dim1[15:0]
    g1[3] = (int)((128u << 16) | ((unsigned)NN >> 16)); // tile_dim0=128 | dim1 hi
    g1[4] = 8;                                    // tile_dim1 = 8 indices
    g1[5] = 128;                                  // tensor_dim0_stride = 128
    g1[6] = 0;
    g1[7] = 0;
    v8i gx;                                       // all 8 row indices packed
    gx[0] = idx_lo[0]; gx[1] = idx_lo[1]; gx[2] = idx_lo[2]; gx[3] = idx_lo[3];
    gx[4] = idx_hi[0]; gx[5] = idx_hi[1]; gx[6] = idx_hi[2]; gx[7] = idx_hi[3];
    __builtin_amdgcn_tensor_load_to_lds(g0, g1, idx_lo, idx_hi, gx, /*cpol=*/0);
}

// Gather 16 rows into a 16x128 f32 LDS tile via two TDM gather ops.
// myRow holds this lane's row index (lanes l and l+16 duplicate row l&15).
__device__ __forceinline__ void tdm_gather16(float* ldsbuf, const float* tab,
                                             int myRow) {
    unsigned int base = lds_byte_off(ldsbuf);
    v4i i0, i1;
    i0[0] = __builtin_amdgcn_readlane(myRow, 0);
    i0[1] = __builtin_amdgcn_readlane(myRow, 1);
    i0[2] = __builtin_amdgcn_readlane(myRow, 2);
    i0[3] = __builtin_amdgcn_readlane(myRow, 3);
    i1[0] = __builtin_amdgcn_readlane(myRow, 4);
    i1[1] = __builtin_amdgcn_readlane(myRow, 5);
    i1[2] = __builtin_amdgcn_readlane(myRow, 6);
    i1[3] = __builtin_amdgcn_readlane(myRow, 7);
    tdm_gather8(base, tab, i0, i1);
    i0[0] = __builtin_amdgcn_readlane(myRow, 8);
    i0[1] = __builtin_amdgcn_readlane(myRow, 9);
    i0[2] = __builtin_amdgcn_readlane(myRow, 10);
    i0[3] = __builtin_amdgcn_readlane(myRow, 11);
    i1[0] = __builtin_amdgcn_readlane(myRow, 12);
    i1[1] = __builtin_amdgcn_readlane(myRow, 13);
    i1[2] = __builtin_amdgcn_readlane(myRow, 14);
    i1[3] = __builtin_amdgcn_readlane(myRow, 15);
    tdm_gather8(base + 8u * DD * 4u, tab, i0, i1);
}

// Full-precision f32 WMMA: accumulate Center(16x128) x Other^T(128x16) over
// 32 chunks of K=4 with V_WMMA_F32_16X16X4_F32, return this lane's diagonal
// element (lanes 0-7: rows 0-7 ; lanes 24-31: rows 8-15).
__device__ __forceinline__ float diag_dot128(const float* lds_a,
                                             const float* lds_b, int lane) {
    v8f acc = {0.f, 0.f, 0.f, 0.f, 0.f, 0.f, 0.f, 0.f};
    const int off = (lane & 15) * DD + ((lane >> 4) << 1);
#pragma unroll
    for (int c = 0; c < 32; ++c) {
        v2f a = *(const v2f*)(lds_a + off + c * 4);
        v2f b = *(const v2f*)(lds_b + off + c * 4);
        acc = __builtin_amdgcn_wmma_f32_16x16x4_f32(
            false, a, false, b, (short)0, acc, false, false);
    }
    int comp = (lane < 8) ? lane : (lane - 24);
    comp &= 7;
    float d = (comp == 0) ? acc[0] :
              (comp == 1) ? acc[1] :
              (comp == 2) ? acc[2] :
              (comp == 3) ? acc[3] :
              (comp == 4) ? acc[4] :
              (comp == 5) ? acc[5] :
              (comp == 6) ? acc[6] : acc[7];
    return d;
}

__global__ void __launch_bounds__(32)
sgns_zero_kernel(float* out) {
    if (threadIdx.x == 0 && blockIdx.x == 0) out[0] = 0.0f;
}

__global__ void __launch_bounds__(32)
sgns_wmma_kernel(const int* __restrict__ countP,
                 const int* __restrict__ shuf,       // (V, TOT)
                 const int* __restrict__ nodes_idx,  // (V, TOT)
                 const int* __restrict__ neigh_idx,  // (V, TOT)
                 const int* __restrict__ neg1,       // (V, B, K)
                 const int* __restrict__ neg2,       // (V, V, B, K)
                 const int* __restrict__ neg3,       // (V, V, B, K)
                 const float* __restrict__ node_W,   // (V, N, D)
                 const float* __restrict__ neigh_W,  // (V, N, D)
                 const float* __restrict__ hyp1,
                 const float* __restrict__ hyp2,
                 float* __restrict__ out)
{
    __shared__ float lds_c[16 * DD];   // center tile (8 KB)
    __shared__ float bufA[16 * DD];    // ctx / neg double buffer (8 KB)
    __shared__ float bufB[16 * DD];    // neg double buffer (8 KB)

    const int lane = threadIdx.x;
    const int wid  = blockIdx.x;
    const int t    = wid / TILES;      // term 0..14
    const int tile = wid % TILES;

    // term -> (kind, i, j); cost2/3 pair order: (j,i) for j in 0..2, i != j
    int kind, ii, jj;
    if (t < 3) { kind = 0; ii = t; jj = t; }
    else {
        kind = (t < 9) ? 1 : 2;
        int p = (t < 9) ? (t - 3) : (t - 9);
        jj = p >> 1;
        int r = p & 1;
        ii = (r == 0) ? ((jj == 0) ? 1 : 0) : ((jj == 2) ? 1 : 2);
    }

    const int count = countP[0];
    const int b     = tile * 16 + (lane & 15);   // lanes l, l+16 duplicate row
    const int sh    = shuf[ii * TOT + count + b];
    const int nsel  = nodes_idx[ii * TOT + sh];
    const int gsel  = neigh_idx[ii * TOT + sh];

    const float* centerTab = node_W + (size_t)ii * NN * DD;   // node_e[i]
    const float* ctxTab;  int ctxRow;
    const float* negTab;  const int* negIdx;
    float w;
    if (kind == 0) {            // cost1[i]
        ctxTab = neigh_W + (size_t)ii * NN * DD;  ctxRow = gsel;
        negTab = ctxTab;
        negIdx = neg1 + ((size_t)ii * BB + b) * KK;
        w = 1.0f / 9.0f;
    } else if (kind == 1) {     // cost2[j,i]
        ctxTab = node_W + (size_t)jj * NN * DD;   ctxRow = nsel;
        negTab = ctxTab;
        negIdx = neg2 + (((size_t)jj * VV + ii) * BB + b) * KK;
        w = hyp1[0] / 18.0f;
    } else {                    // cost3[j,i]
        ctxTab = neigh_W + (size_t)jj * NN * DD;  ctxRow = gsel;
        negTab = ctxTab;
        negIdx = neg3 + (((size_t)jj * VV + ii) * BB + b) * KK;
        w = hyp2[0] / 18.0f;
    }

    const bool valid = (lane < 8) || (lane >= 24);  // lanes owning diag elems

    // --- pipelined TDM gathers + WMMA products --------------------------
    tdm_gather16(lds_c, centerTab, nsel);        // 2 TDM ops
    tdm_gather16(bufA,  ctxTab,    ctxRow);      // 2 TDM ops
    __builtin_amdgcn_s_wait_tensorcnt(0);

    int ni0 = negIdx[0];
    tdm_gather16(bufB, negTab, ni0);             // prefetch neg0 (2 ops)

    float lsum = 0.0f;
    {   // positive term
        float d = diag_dot128(lds_c, bufA, lane);
        lsum += valid ? log_sigmoid(d) : 0.0f;
    }

    for (int k = 0; k < KK; ++k) {
        float* cur = (k & 1) ? bufA : bufB;      // where neg_k landed
        float* nxt = (k & 1) ? bufB : bufA;
        if (k + 1 < KK) {
            int nik = negIdx[k + 1];
            tdm_gather16(nxt, negTab, nik);      // issue ahead (<=4 in flight)
            __builtin_amdgcn_s_wait_tensorcnt(2);// neg_k's 2 ops complete
        } else {
            __builtin_amdgcn_s_wait_tensorcnt(0);
        }
        float d = diag_dot128(lds_c, cur, lane);
        lsum += valid ? log_sigmoid(-d) : 0.0f;
    }

    // wave32 reduction
#pragma unroll
    for (int s = 16; s > 0; s >>= 1) lsum += __shfl_down(lsum, s, 32);

    if (lane == 0) atomicAdd(out, -(w / (float)BB) * lsum);
}

extern "C" void kernel_launch(void* const* d_in, const int* in_sizes, int n_in,
                              void* d_out, int out_size, void* d_ws, size_t ws_size,
                              hipStream_t stream) {
    (void)in_sizes; (void)n_in; (void)out_size; (void)d_ws; (void)ws_size;
    const int*   countP    = (const int*)d_in[0];
    const int*   shuf      = (const int*)d_in[1];
    const int*   nodes_idx = (const int*)d_in[2];
    const int*   neigh_idx = (const int*)d_in[3];
    const int*   neg1      = (const int*)d_in[4];
    const int*   neg2      = (const int*)d_in[5];
    const int*   neg3      = (const int*)d_in[6];
    const float* node_W    = (const float*)d_in[7];
    const float* neigh_W   = (const float*)d_in[8];
    const float* hyp1      = (const float*)d_in[9];
    const float* hyp2      = (const float*)d_in[10];
    float* out = (float*)d_out;

    sgns_zero_kernel<<<1, 32, 0, stream>>>(out);
    sgns_wmma_kernel<<<TERMS * TILES, 32, 0, stream>>>(
        countP, shuf, nodes_idx, neigh_idx, neg1, neg2, neg3,
        node_W, neigh_W, hyp1, hyp2, out);
}